// SNNHead_15083925144324
// MI455X (gfx1250) — compile-verified
//
#include <hip/hip_runtime.h>

// ---------------------------------------------------------------------------
// SNN head for MI455X (gfx1250):
//   k1: 1x1 conv as FP32 WMMA GEMM (M=classes padded to 32, N=pixels, K=256)
//       + fused 10-step leaky-integrate-fire recurrence -> mid[B,21,64,64] in ws
//   k2: x4 bilinear upsample (half-pixel) -> out[B,21,256,256]
// Memory-bound design: x streamed once (64 MiB), mid stays L2-resident.
// ---------------------------------------------------------------------------

typedef __attribute__((ext_vector_type(2))) float v2f;
typedef __attribute__((ext_vector_type(8))) float v8f;

#define HW        4096      // 64*64
#define C_IN      256
#define K_CLS     21
#define M_PAD     32
#define LDS_STRIDE 260      // 256 + 4 pad -> distinct LDS banks across class lanes
#define NSTEPS    10

__global__ __launch_bounds__(256) void snn_conv_lif_kernel(
    const float* __restrict__ x,      // [16,256,64,64]
    const float* __restrict__ w,      // [21,256]
    const float* __restrict__ beta,   // [1]
    float* __restrict__ mid)          // [16,21,64,64]
{
    __shared__ float wlds[M_PAD * LDS_STRIDE];

    const int tid = threadIdx.x;

    // Stage zero-padded transposed-access weights into LDS: [class(32)][k(256)]
    for (int i = tid; i < M_PAD * C_IN; i += 256) {
        int cls = i >> 8;        // i / 256
        int k   = i & 255;
        float v = (cls < K_CLS) ? w[cls * C_IN + k] : 0.0f;
        wlds[cls * LDS_STRIDE + k] = v;
    }
    __syncthreads();

    const int bid    = blockIdx.x;          // 128 blocks
    const int b      = bid >> 3;            // batch
    const int hwBase = (bid & 7) << 9;      // 512 pixels per block
    const int wave   = tid >> 5;
    const int lane   = tid & 31;
    const int pix0   = hwBase + wave * 64;  // 64 pixels per wave
    const int laneLo = lane & 15;
    const int laneHi = lane >> 4;           // 0 or 1

    const float* xb = x + (size_t)b * C_IN * HW;

    v8f acc[2][4];
    const v8f vzero = {0.f, 0.f, 0.f, 0.f, 0.f, 0.f, 0.f, 0.f};
#pragma unroll
    for (int mt = 0; mt < 2; ++mt)
#pragma unroll
        for (int nt = 0; nt < 4; ++nt)
            acc[mt][nt] = vzero;

    // K loop: V_WMMA_F32_16X16X4_F32, 4 channels per step.
    // A 16x4 layout: lane%16 = M(row/class), VGPR0/1 hold K = 2*(lane/16)+{0,1}
    // B 4x16 layout (assumed symmetric): lane%16 = N(pixel), same K pairing.
    for (int k0 = 0; k0 < C_IN; k0 += 4) {
        const int krow = k0 + 2 * laneHi;

        v2f afrag[2];
#pragma unroll
        for (int mt = 0; mt < 2; ++mt) {
            const float* ap = &wlds[(mt * 16 + laneLo) * LDS_STRIDE + krow];
            afrag[mt] = *(const v2f*)ap;     // ds_load_b64 (8B aligned: krow even)
        }

#pragma unroll
        for (int nt = 0; nt < 4; ++nt) {
            const int pix = pix0 + nt * 16 + laneLo;
            v2f bfrag;
            bfrag.x = xb[(size_t)(krow + 0) * HW + pix];
            bfrag.y = xb[(size_t)(krow + 1) * HW + pix];
            acc[0][nt] = __builtin_amdgcn_wmma_f32_16x16x4_f32(
                false, afrag[0], false, bfrag, (short)0, acc[0][nt], false, false);
            acc[1][nt] = __builtin_amdgcn_wmma_f32_16x16x4_f32(
                false, afrag[1], false, bfrag, (short)0, acc[1][nt], false, false);
        }
    }

    float bclip = beta[0];
    bclip = fminf(fmaxf(bclip, 0.0f), 1.0f);

    // Fused LIF recurrence (cur constant per element) + store spike rate.
    // C/D layout: VGPR r -> class = mt*16 + r + 8*laneHi; lane%16 -> pixel.
#pragma unroll
    for (int mt = 0; mt < 2; ++mt) {
#pragma unroll
        for (int r = 0; r < 8; ++r) {
            const int cls = mt * 16 + r + 8 * laneHi;
            if (cls < K_CLS) {
                float* dst = mid + ((size_t)b * K_CLS + cls) * HW;
#pragma unroll
                for (int nt = 0; nt < 4; ++nt) {
                    const float cur = acc[mt][nt][r];
                    float mem = 0.0f, s = 0.0f;
#pragma unroll
                    for (int t = 0; t < NSTEPS; ++t) {
                        const float reset = (mem > 1.0f) ? 1.0f : 0.0f;
                        mem = bclip * mem + cur - reset;
                        s += (mem > 1.0f) ? 1.0f : 0.0f;
                    }
                    dst[pix0 + nt * 16 + laneLo] = s * 0.1f;
                }
            }
        }
    }
}

__global__ __launch_bounds__(256) void upsample4x_bilinear_kernel(
    const float* __restrict__ mid,    // [336,64,64]  (B*K planes)
    float* __restrict__ out)          // [336,256,256]
{
    const int idx   = blockIdx.x * 256 + threadIdx.x;
    const int ox    = idx & 255;
    const int oy    = (idx >> 8) & 255;
    const int plane = idx >> 16;                 // b*21 + cls, 0..335
    const float* src = mid + (size_t)plane * HW;

    const float sx = (ox + 0.5f) * 0.25f - 0.5f;
    const float sy = (oy + 0.5f) * 0.25f - 0.5f;
    const float fx0 = floorf(sx), fy0 = floorf(sy);
    const float wx = sx - fx0, wy = sy - fy0;
    const int x0 = (int)fx0, y0 = (int)fy0;
    const int x0c = max(x0, 0),     y0c = max(y0, 0);
    const int x1c = min(x0 + 1, 63), y1c = min(y0 + 1, 63);

    const float v00 = src[y0c * 64 + x0c];
    const float v01 = src[y0c * 64 + x1c];
    const float v10 = src[y1c * 64 + x0c];
    const float v11 = src[y1c * 64 + x1c];

    const float top = v00 + wx * (v01 - v00);
    const float bot = v10 + wx * (v11 - v10);
    out[idx] = top + wy * (bot - top);
}

extern "C" void kernel_launch(void* const* d_in, const int* in_sizes, int n_in,
                              void* d_out, int out_size, void* d_ws, size_t ws_size,
                              hipStream_t stream) {
    const float* x    = (const float*)d_in[0];   // [16,256,64,64]
    const float* w    = (const float*)d_in[1];   // [21,256]
    const float* beta = (const float*)d_in[2];   // [1]
    float* mid = (float*)d_ws;                   // needs 16*21*4096*4 = 5.25 MiB
    float* out = (float*)d_out;                  // [16,21,256,256]

    // 16 batches * 8 blocks/batch; 256 threads = 8 waves, 64 pixels/wave
    snn_conv_lif_kernel<<<128, 256, 0, stream>>>(x, w, beta, mid);

    // 16*21*256*256 / 256 = 86016 blocks
    upsample4x_bilinear_kernel<<<86016, 256, 0, stream>>>(mid, out);
}